// KF_1271310319800
// MI455X (gfx1250) — compile-verified
//
#include <hip/hip_runtime.h>
#include <math.h>

#if defined(__HIP_DEVICE_COMPILE__) && !__has_builtin(__builtin_amdgcn_wmma_f32_16x16x4_f32)
#error "__builtin_amdgcn_wmma_f32_16x16x4_f32 not available on this device toolchain"
#endif

typedef __attribute__((ext_vector_type(2))) float v2f;
typedef __attribute__((ext_vector_type(8))) float v8f;

#define N_TRAJ 256
#define T_OBS  1000
#define NS     10
#define LD     17   // LDS row stride (floats); gcd(17,64)=1 -> conflict-free columns

static constexpr size_t TRAJ_ELEMS = (size_t)N_TRAJ * (T_OBS + 1) * NS;        // 2,562,560
static constexpr size_t PK_ELEMS   = (size_t)N_TRAJ * (T_OBS + 1) * NS * NS;   // 25,625,600
static constexpr size_t MSE_OFF    = TRAJ_ELEMS + PK_ELEMS;                    // 28,188,160

// ---------------------------------------------------------------------------
// Wave-cooperative 16x16x16 f32 GEMM: D = A*B using V_WMMA_F32_16X16X4_F32.
// A, B, D are 16x16 tiles in LDS (row-major, stride LD). Optionally also
// stores D transposed. All 32 lanes must be active (WMMA EXEC requirement).
// ---------------------------------------------------------------------------
template <bool STORE_T>
__device__ __forceinline__ void mm16(const float (&A)[16][LD],
                                     const float (&B)[16][LD],
                                     float (&D)[16][LD],
                                     float (*DT)[LD],
                                     int lane)
{
    const int nM    = lane & 15;          // column for B/D fragments, row for A
    const int half2 = (lane >> 4) << 1;   // 0 or 2 : K-offset of this half-wave
    v8f c = {0.f, 0.f, 0.f, 0.f, 0.f, 0.f, 0.f, 0.f};
#pragma unroll
    for (int kk = 0; kk < 16; kk += 4) {
        v2f a, b;
        // A 16x4 fragment: VGPR0 -> K = kk+half2, VGPR1 -> K = kk+half2+1
        a.x = A[nM][kk + half2];
        a.y = A[nM][kk + half2 + 1];
        // B 4x16 fragment: VGPR0 -> row kk+half2, VGPR1 -> row kk+half2+1
        b.x = B[kk + half2][nM];
        b.y = B[kk + half2 + 1][nM];
        c = __builtin_amdgcn_wmma_f32_16x16x4_f32(false, a, false, b,
                                                  (short)0, c, false, false);
    }
    const int mBase = (lane >> 4) * 8;    // C/D: VGPR v holds row v (+8 for hi half)
#pragma unroll
    for (int v = 0; v < 8; ++v) {
        D[mBase + v][nM] = c[v];
        if (STORE_T) DT[nM][mBase + v] = c[v];
    }
}

// ---------------------------------------------------------------------------
// One wave32 per trajectory: 1000 sequential covariance-form KF steps.
// ---------------------------------------------------------------------------
__global__ __launch_bounds__(32) void kf_kernel(const float* __restrict__ Y,
                                                const float* __restrict__ F,
                                                const float* __restrict__ H,
                                                const float* __restrict__ Q,
                                                const float* __restrict__ R,
                                                float* __restrict__ traj,
                                                float* __restrict__ Pk)
{
    struct SM {
        float Fm[16][LD], Ft[16][LD], Hm[16][LD], Ht[16][LD];
        float Qm[16][LD], Rm[16][LD];
        float P[16][LD];            // P_pos
        float T1[16][LD];           // F*P
        float Pn[16][LD];           // P_neg
        float PHt[16][LD], PHtT[16][LD];
        float Lc[16][LD];           // S, then chol(S) in place (lower)
        float Kk[16][LD];           // Kalman gain (padded with zeros)
        float T2[16][LD];           // K * PHt^T
        float x[16], xn[16], innov[16], yv[16];
    };
    __shared__ SM sm;

    const int lane = threadIdx.x;
    const int b    = blockIdx.x;

    // Zero all of LDS (pad regions must stay zero for the padded 16x16 GEMMs).
    {
        float* f = (float*)&sm;
        const int total = (int)(sizeof(SM) / sizeof(float));
        for (int i = lane; i < total; i += 32) f[i] = 0.f;
    }
    __syncthreads();

    // Load constant matrices (10x10) and their transposes into padded tiles.
    for (int idx = lane; idx < NS * NS; idx += 32) {
        const int r = idx / NS, c = idx % NS;
        const float fv = F[idx], hv = H[idx];
        sm.Fm[r][c] = fv;  sm.Ft[c][r] = fv;
        sm.Hm[r][c] = hv;  sm.Ht[c][r] = hv;
        sm.Qm[r][c] = Q[idx];
        sm.Rm[r][c] = R[idx];
    }
    if (lane < NS) sm.P[lane][lane] = 1.f;   // P0 = I, x0 = 0 (already zeroed)

    const float* Yb    = Y    + (size_t)b * T_OBS * NS;
    float*       trajb = traj + (size_t)b * (T_OBS + 1) * NS;
    float*       Pkb   = Pk   + (size_t)b * (T_OBS + 1) * NS * NS;

    if (lane < NS) trajb[lane] = 0.f;
    for (int i = lane; i < NS * NS; i += 32) Pkb[i] = 0.f;
    __syncthreads();

#pragma unroll 1
    for (int t = 0; t < T_OBS; ++t) {
        if (t + 1 < T_OBS) __builtin_prefetch(&Yb[(t + 1) * NS], 0, 0);

        // y_t and x_neg = F x_pos
        if (lane < NS) {
            sm.yv[lane] = Yb[t * NS + lane];
            float s = 0.f;
#pragma unroll
            for (int j = 0; j < NS; ++j) s += sm.Fm[lane][j] * sm.x[j];
            sm.xn[lane] = s;
        }
        __syncthreads();
        // innovation = y - H x_neg
        if (lane < NS) {
            float s = 0.f;
#pragma unroll
            for (int j = 0; j < NS; ++j) s += sm.Hm[lane][j] * sm.xn[j];
            sm.innov[lane] = sm.yv[lane] - s;
        }

        // P_neg = F P F^T + Q
        mm16<false>(sm.Fm, sm.P, sm.T1, nullptr, lane);
        __syncthreads();
        mm16<false>(sm.T1, sm.Ft, sm.Pn, nullptr, lane);
        __syncthreads();
        for (int i = lane; i < 256; i += 32) {
            const int r = i >> 4, c = i & 15;
            sm.Pn[r][c] += sm.Qm[r][c];
        }
        __syncthreads();

        // PHt = P_neg H^T  (also keep its transpose = H P_neg)
        mm16<true>(sm.Pn, sm.Ht, sm.PHt, sm.PHtT, lane);
        __syncthreads();

        // S = H PHt + R  (into Lc)
        mm16<false>(sm.Hm, sm.PHt, sm.Lc, nullptr, lane);
        __syncthreads();
        for (int i = lane; i < 256; i += 32) {
            const int r = i >> 4, c = i & 15;
            sm.Lc[r][c] += sm.Rm[r][c];
        }
        __syncthreads();

        // Cholesky S = L L^T in place (lower triangle of Lc, 10x10 active)
#pragma unroll 1
        for (int j = 0; j < NS; ++j) {
            if (lane == 0) {
                float s = sm.Lc[j][j];
                for (int k = 0; k < j; ++k) s -= sm.Lc[j][k] * sm.Lc[j][k];
                sm.Lc[j][j] = sqrtf(fmaxf(s, 1e-30f));
            }
            __syncthreads();
            if (lane > j && lane < NS) {
                float s = sm.Lc[lane][j];
                for (int k = 0; k < j; ++k) s -= sm.Lc[lane][k] * sm.Lc[j][k];
                sm.Lc[lane][j] = s / sm.Lc[j][j];
            }
            __syncthreads();
        }

        // K = PHt (L L^T)^{-1} : two triangular solves, lanes = rows of PHt
        if (lane < NS) {
            float w[NS];
#pragma unroll
            for (int j = 0; j < NS; ++j) {        // W = PHt L^{-T} (forward)
                float s = sm.PHt[lane][j];
#pragma unroll
                for (int k = 0; k < j; ++k) s -= w[k] * sm.Lc[j][k];
                w[j] = s / sm.Lc[j][j];
            }
            float kv[NS];
#pragma unroll
            for (int j = NS - 1; j >= 0; --j) {   // K = W L^{-1} (backward)
                float s = w[j];
#pragma unroll
                for (int k = j + 1; k < NS; ++k) s -= kv[k] * sm.Lc[k][j];
                kv[j] = s / sm.Lc[j][j];
            }
#pragma unroll
            for (int j = 0; j < NS; ++j) sm.Kk[lane][j] = kv[j];
        }
        __syncthreads();

        // P_pos = P_neg - K (PHt)^T ; x_pos = x_neg + K innov
        mm16<false>(sm.Kk, sm.PHtT, sm.T2, nullptr, lane);
        __syncthreads();
        for (int i = lane; i < 256; i += 32) {
            const int r = i >> 4, c = i & 15;
            sm.P[r][c] = sm.Pn[r][c] - sm.T2[r][c];
        }
        if (lane < NS) {
            float s = sm.xn[lane];
#pragma unroll
            for (int j = 0; j < NS; ++j) s += sm.Kk[lane][j] * sm.innov[j];
            sm.x[lane] = s;
        }
        __syncthreads();

        // Emit outputs for this step
        if (lane < NS) trajb[(t + 1) * NS + lane] = sm.x[lane];
        for (int i = lane; i < NS * NS; i += 32)
            Pkb[(t + 1) * NS * NS + i] = sm.P[i / NS][i % NS];
        __syncthreads();
    }
}

// ---------------------------------------------------------------------------
// MSE reduction: per-trajectory 10*log10(mean sq err), then average in dB.
// ---------------------------------------------------------------------------
__global__ __launch_bounds__(256) void mse_traj_kernel(const float* __restrict__ X,
                                                       const float* __restrict__ traj,
                                                       float* __restrict__ ws)
{
    __shared__ float red[256];
    const int b = blockIdx.x, tid = threadIdx.x;
    const float* xb = X    + (size_t)b * (T_OBS + 1) * NS;
    const float* tb = traj + (size_t)b * (T_OBS + 1) * NS;
    float s = 0.f;
    for (int i = tid; i < T_OBS * NS; i += 256) {
        const float d = xb[NS + i] - tb[NS + i];   // skip t = 0 row
        s += d * d;
    }
    red[tid] = s;
    __syncthreads();
    for (int off = 128; off > 0; off >>= 1) {
        if (tid < off) red[tid] += red[tid + off];
        __syncthreads();
    }
    if (tid == 0) ws[b] = 10.f * log10f(red[0] * (1.f / (float)(T_OBS * NS)));
}

__global__ __launch_bounds__(256) void mse_avg_kernel(const float* __restrict__ ws,
                                                      float* __restrict__ out)
{
    __shared__ float red[256];
    const int tid = threadIdx.x;
    red[tid] = ws[tid];
    __syncthreads();
    for (int off = 128; off > 0; off >>= 1) {
        if (tid < off) red[tid] += red[tid + off];
        __syncthreads();
    }
    if (tid == 0) out[0] = red[0] * (1.f / (float)N_TRAJ);
}

// ---------------------------------------------------------------------------
extern "C" void kernel_launch(void* const* d_in, const int* in_sizes, int n_in,
                              void* d_out, int out_size, void* d_ws, size_t ws_size,
                              hipStream_t stream)
{
    const float* X = (const float*)d_in[0];
    const float* Y = (const float*)d_in[1];
    const float* F = (const float*)d_in[2];
    const float* H = (const float*)d_in[3];
    const float* Q = (const float*)d_in[4];
    const float* R = (const float*)d_in[5];

    float* out  = (float*)d_out;
    float* traj = out;                 // [256, 1001, 10]
    float* Pk   = out + TRAJ_ELEMS;    // [256, 1001, 10, 10]
    float* mse  = out + MSE_OFF;       // scalar
    float* ws   = (float*)d_ws;        // 256 floats of scratch

    kf_kernel<<<dim3(N_TRAJ), dim3(32), 0, stream>>>(Y, F, H, Q, R, traj, Pk);
    mse_traj_kernel<<<dim3(N_TRAJ), dim3(256), 0, stream>>>(X, traj, ws);
    mse_avg_kernel<<<dim3(1), dim3(256), 0, stream>>>(ws, mse);
}